// GIN_78975858638933
// MI455X (gfx1250) — compile-verified
//
#include <hip/hip_runtime.h>

typedef __attribute__((ext_vector_type(2))) float v2f;
typedef __attribute__((ext_vector_type(8))) float v8f;

#define HIDDEN 64
#define XS_STRIDE 66   // even pad: keeps 8B alignment for v2f LDS reads, spreads banks

// ---------------- utility ----------------
__global__ void zero_kernel(float* __restrict__ p, long long n) {
  long long i = (long long)blockIdx.x * blockDim.x + threadIdx.x;
  if (i < n) p[i] = 0.0f;
}

// ---------------- edge kernels ----------------
__global__ void edge_deg_kernel(const int* __restrict__ dst, float* __restrict__ deg, int E) {
  int e = blockIdx.x * blockDim.x + threadIdx.x;
  if (e < E) atomicAdd(&deg[dst[e]], 1.0f);
}

__global__ void edge_agg1_kernel(const int* __restrict__ src, const int* __restrict__ dst,
                                 const float* __restrict__ deg, float* __restrict__ t1, int E) {
  int e = blockIdx.x * blockDim.x + threadIdx.x;
  if (e < E) atomicAdd(&t1[dst[e]], deg[src[e]]);
}

// 16 lanes per edge, each lane moves one float4 (256B coalesced gather per edge)
__global__ void edge_agg64_kernel(const int* __restrict__ src, const int* __restrict__ dst,
                                  const float* __restrict__ H, float* __restrict__ T, int E) {
  long long t = (long long)blockIdx.x * blockDim.x + threadIdx.x;
  int e = (int)(t >> 4);
  if (e < E) {
    int c4 = ((int)t & 15) << 2;
    int s = src[e], d = dst[e];
    const float4 v = *(const float4*)(H + (size_t)s * HIDDEN + c4);
    float* o = T + (size_t)d * HIDDEN + c4;
    atomicAdd(o + 0, v.x);
    atomicAdd(o + 1, v.y);
    atomicAdd(o + 2, v.z);
    atomicAdd(o + 3, v.w);
  }
}

// ---------------- layer-0 rank-1 first MLP stage ----------------
// X[n][c] = relu((deg[n] + t1[n]) * W1[c] + b1[c])
__global__ void layer0_x_kernel(const float* __restrict__ deg, const float* __restrict__ t1,
                                const float* __restrict__ W1, const float* __restrict__ b1,
                                float* __restrict__ X, int N) {
  int i = blockIdx.x * blockDim.x + threadIdx.x;  // over N*64 (6.4M, fits int)
  if (i < N * HIDDEN) {
    int n = i >> 6, c = i & 63;
    float h = deg[n] + t1[n];
    float v = fmaf(h, W1[c], b1[c]);
    X[i] = v > 0.0f ? v : 0.0f;
  }
}

// ---------------- fused WMMA MLP: Y = relu(A@W1+b1)@W2 + b2, plus BN stats ----------------
// If W1 != nullptr, A1 MUST also be non-null (fused GIN self-loop add A0+A1).
// If W1 == nullptr, A0 already holds X (first stage precomputed).
// One wave per 16-row tile; 8 waves (256 threads) per block -> 128 rows/block.
__global__ __launch_bounds__(256) void mlp_wmma_kernel(
    const float* __restrict__ A0, const float* __restrict__ A1,
    const float* __restrict__ W1, const float* __restrict__ b1,
    const float* __restrict__ W2, const float* __restrict__ b2,
    float* __restrict__ Y, float* __restrict__ stats, int N) {
  __shared__ float Xs[8][16 * XS_STRIDE];
  __shared__ float ssum[HIDDEN];
  __shared__ float ssq[HIDDEN];

  const int tid = threadIdx.x;
  const int wave = tid >> 5;   // wave32
  const int lane = tid & 31;
  const int lo = lane & 15;
  const int hi = lane >> 4;    // half-wave selects K sub-pair {0,1} vs {2,3}
  const int rowBase = (blockIdx.x * 8 + wave) * 16;

  if (tid < HIDDEN) { ssum[tid] = 0.0f; ssq[tid] = 0.0f; }
  __syncthreads();

  float* xs = Xs[wave];

  if (W1 != nullptr) {
    // ---- first GEMM: X = relu((A0+A1) @ W1 + b1) ----
    v8f acc[4] = {};
    int ar = rowBase + lo;
    if (ar >= N) ar = N - 1;  // clamp; tail rows produce garbage that is never stored
    const float* arow0 = A0 + (size_t)ar * HIDDEN;
    const float* arow1 = A1 + (size_t)ar * HIDDEN;
    for (int s = 0; s < 16; ++s) {
      const int ka = s * 4 + hi * 2;        // even -> 8B-aligned v2f load
      const v2f a0 = *(const v2f*)(arow0 + ka);
      const v2f a1 = *(const v2f*)(arow1 + ka);
      const v2f a = a0 + a1;
      for (int t = 0; t < 4; ++t) {
        const int col = t * 16 + lo;
        v2f b;
        b.x = W1[ka * HIDDEN + col];
        b.y = W1[(ka + 1) * HIDDEN + col];
        acc[t] = __builtin_amdgcn_wmma_f32_16x16x4_f32(
            false, a, false, b, (short)0, acc[t], false, false);
      }
    }
    // bias + relu, stage X tile into LDS (row-major, padded stride)
    for (int t = 0; t < 4; ++t) {
      const int col = t * 16 + lo;
      const float bias = b1[col];
      for (int j = 0; j < 8; ++j) {
        float v = acc[t][j] + bias;
        v = v > 0.0f ? v : 0.0f;
        xs[(j + hi * 8) * XS_STRIDE + col] = v;
      }
    }
  } else {
    // X precomputed in A0: stage the wave's 16x64 tile into LDS
    for (int idx = lane; idx < 16 * HIDDEN; idx += 32) {
      int r = idx >> 6, c = idx & 63;
      int ar = rowBase + r;
      if (ar >= N) ar = N - 1;
      xs[r * XS_STRIDE + c] = A0[(size_t)ar * HIDDEN + c];
    }
  }

  // ---- second GEMM: Y = X @ W2 + b2 ----
  v8f acc2[4] = {};
  for (int s = 0; s < 16; ++s) {
    const int ka = s * 4 + hi * 2;
    const v2f a = *(const v2f*)(xs + lo * XS_STRIDE + ka);  // 8B-aligned (stride 66)
    for (int t = 0; t < 4; ++t) {
      const int col = t * 16 + lo;
      v2f b;
      b.x = W2[ka * HIDDEN + col];
      b.y = W2[(ka + 1) * HIDDEN + col];
      acc2[t] = __builtin_amdgcn_wmma_f32_16x16x4_f32(
          false, a, false, b, (short)0, acc2[t], false, false);
    }
  }

  // bias, store Y, accumulate per-column BN stats (block-local LDS first)
  for (int t = 0; t < 4; ++t) {
    const int col = t * 16 + lo;
    const float bias = b2[col];
    float ps = 0.0f, pq = 0.0f;
    for (int j = 0; j < 8; ++j) {
      const int row = rowBase + j + hi * 8;
      const float v = acc2[t][j] + bias;
      if (row < N) {
        Y[(size_t)row * HIDDEN + col] = v;
        ps += v;
        pq += v * v;
      }
    }
    atomicAdd(&ssum[col], ps);
    atomicAdd(&ssq[col], pq);
  }
  __syncthreads();
  if (tid < HIDDEN) {
    atomicAdd(&stats[tid], ssum[tid]);
    atomicAdd(&stats[HIDDEN + tid], ssq[tid]);
  }
}

// ---------------- BN finalize + apply ----------------
__global__ void bn_finalize_kernel(const float* __restrict__ stats,
                                   const float* __restrict__ gamma, const float* __restrict__ beta,
                                   float* __restrict__ ss, float invN) {
  int c = threadIdx.x;
  if (c < HIDDEN) {
    float mean = stats[c] * invN;
    float var = stats[HIDDEN + c] * invN - mean * mean;
    float sc = gamma[c] * rsqrtf(var + 1e-5f);
    ss[c] = sc;
    ss[HIDDEN + c] = beta[c] - mean * sc;
  }
}

__global__ void bn_apply_kernel(const float* __restrict__ Y, const float* __restrict__ ss,
                                float* __restrict__ out, int N) {
  int i = blockIdx.x * blockDim.x + threadIdx.x;  // over N*16 float4 groups
  if (i < N * 16) {
    const int c4 = (i & 15) << 2;
    const float4 y = *(const float4*)(Y + (size_t)i * 4);
    float4 o;
    o.x = fmaxf(0.0f, fmaf(y.x, ss[c4 + 0], ss[HIDDEN + c4 + 0]));
    o.y = fmaxf(0.0f, fmaf(y.y, ss[c4 + 1], ss[HIDDEN + c4 + 1]));
    o.z = fmaxf(0.0f, fmaf(y.z, ss[c4 + 2], ss[HIDDEN + c4 + 2]));
    o.w = fmaxf(0.0f, fmaf(y.w, ss[c4 + 3], ss[HIDDEN + c4 + 3]));
    *(float4*)(out + (size_t)i * 4) = o;
  }
}

// ---------------- host ----------------
extern "C" void kernel_launch(void* const* d_in, const int* in_sizes, int n_in,
                              void* d_out, int out_size, void* d_ws, size_t ws_size,
                              hipStream_t stream) {
  const int* src = (const int*)d_in[0];
  const int* dst = (const int*)d_in[1];
  const float* W1_0 = (const float*)d_in[2];
  const float* b1_0 = (const float*)d_in[3];
  const float* W2_0 = (const float*)d_in[4];
  const float* b2_0 = (const float*)d_in[5];
  const float* Ws1 = (const float*)d_in[6];
  const float* bs1 = (const float*)d_in[7];
  const float* Ws2 = (const float*)d_in[8];
  const float* bs2 = (const float*)d_in[9];
  const float* gamma = (const float*)d_in[10];
  const float* beta = (const float*)d_in[11];

  const int E = in_sizes[0];
  const int N = out_size / HIDDEN;

  // workspace carve-up (floats): deg[N], t1[N], Y[N*64], H[N*64], T[N*64], stats[128], ss[128]
  float* w = (float*)d_ws;
  float* deg = w;
  float* t1 = deg + N;
  float* Ybuf = t1 + N;
  float* Hbuf = Ybuf + (size_t)N * HIDDEN;
  float* Tbuf = Hbuf + (size_t)N * HIDDEN;
  float* stats = Tbuf + (size_t)N * HIDDEN;
  float* ss = stats + 128;

  const int B = 256;
  const long long n64 = (long long)N * HIDDEN;
  const int mlpBlocks = (N + 127) / 128;

  // degree + scalar aggregation (layer 0 input h0 = deg + sum_in deg[src])
  zero_kernel<<<(int)((2LL * N + B - 1) / B), B, 0, stream>>>(deg, 2LL * N);
  edge_deg_kernel<<<(E + B - 1) / B, B, 0, stream>>>(dst, deg, E);
  edge_agg1_kernel<<<(E + B - 1) / B, B, 0, stream>>>(src, dst, deg, t1, E);

  // layer 0: X = relu(h0 * W1_0 + b1_0) -> Tbuf; Y = X @ W2_0 + b2_0; BN+relu -> Hbuf
  layer0_x_kernel<<<(int)((n64 + B - 1) / B), B, 0, stream>>>(deg, t1, W1_0, b1_0, Tbuf, N);
  zero_kernel<<<1, B, 0, stream>>>(stats, 256);
  mlp_wmma_kernel<<<mlpBlocks, B, 0, stream>>>(Tbuf, nullptr, nullptr, nullptr,
                                               W2_0, b2_0, Ybuf, stats, N);
  bn_finalize_kernel<<<1, 64, 0, stream>>>(stats, gamma, beta, ss, 1.0f / (float)N);
  bn_apply_kernel<<<(N * 16 + B - 1) / B, B, 0, stream>>>(Ybuf, ss, Hbuf, N);

  // layers 1..3
  for (int i = 0; i < 3; ++i) {
    zero_kernel<<<(int)((n64 + B - 1) / B), B, 0, stream>>>(Tbuf, n64);
    long long aggThreads = (long long)E * 16;
    edge_agg64_kernel<<<(int)((aggThreads + B - 1) / B), B, 0, stream>>>(src, dst, Hbuf, Tbuf, E);
    zero_kernel<<<1, B, 0, stream>>>(stats, 256);
    mlp_wmma_kernel<<<mlpBlocks, B, 0, stream>>>(Hbuf, Tbuf,
                                                 Ws1 + (size_t)i * HIDDEN * HIDDEN, bs1 + (size_t)i * HIDDEN,
                                                 Ws2 + (size_t)i * HIDDEN * HIDDEN, bs2 + (size_t)i * HIDDEN,
                                                 Ybuf, stats, N);
    bn_finalize_kernel<<<1, 64, 0, stream>>>(stats, gamma + (size_t)(i + 1) * HIDDEN,
                                             beta + (size_t)(i + 1) * HIDDEN, ss, 1.0f / (float)N);
    float* dest = (i == 2) ? (float*)d_out : Hbuf;
    bn_apply_kernel<<<(N * 16 + B - 1) / B, B, 0, stream>>>(Ybuf, ss, dest, N);
  }
}